// CTCDecode_45990509805818
// MI455X (gfx1250) — compile-verified
//
#include <hip/hip_runtime.h>
#include <hip/hip_bf16.h>
#include <stdint.h>

// Problem constants (from the reference): B=1024, T=512, C=128, blank=C-1.
#define B_ 1024
#define T_ 512
#define C_ 128

typedef float v4f __attribute__((ext_vector_type(4)));

// ---------------------------------------------------------------------------
// Kernel 1: per-(b,t) argmax over C=128 classes.
// One lane per row (row = b*T + t). 32 unrolled nontemporal b128 loads stream
// the 512B row; strict '>' keeps the FIRST occurrence of the max, matching
// jnp.argmax tie-breaking. Output: u8 label per row into workspace.
// Memory-bound: 268 MB probs -> ~11.5us at 23.3 TB/s.
// ---------------------------------------------------------------------------
__global__ __launch_bounds__(256) void ctc_argmax_kernel(
    const float* __restrict__ probs, unsigned char* __restrict__ best) {
  const int gid = blockIdx.x * 256 + threadIdx.x;  // row index b*T + t
  const v4f* row = (const v4f*)(probs + (size_t)gid * C_);
  float maxv = -1.0f;  // probs are uniform [0,1): first element always wins vs -1
  int maxi = 0;
#pragma unroll
  for (int c = 0; c < C_ / 4; ++c) {
    v4f v = __builtin_nontemporal_load(&row[c]);  // streamed once: NT hint
    if (v.x > maxv) { maxv = v.x; maxi = 4 * c + 0; }
    if (v.y > maxv) { maxv = v.y; maxi = 4 * c + 1; }
    if (v.z > maxv) { maxv = v.z; maxi = 4 * c + 2; }
    if (v.w > maxv) { maxv = v.w; maxi = 4 * c + 3; }
  }
  best[gid] = (unsigned char)maxi;
}

// ---------------------------------------------------------------------------
// Kernel 2: per-batch-row CTC collapse + left-pack + table lookup.
// One wave32 per row b. The 512B label row and the first 128B of the lookup table are
// staged into LDS with GLOBAL_LOAD_ASYNC_TO_LDS_B128 (CDNA5 ASYNCcnt path),
// then a 16-step wave scan (ds_bpermute + ballot_w32 prefix) compacts in-place.
// ---------------------------------------------------------------------------
__global__ __launch_bounds__(256) void ctc_compact_kernel(
    const unsigned char* __restrict__ best, const int* __restrict__ table,
    const int* __restrict__ defc_p, float* __restrict__ out) {
  __shared__ __align__(16) unsigned char sBest[8][T_];  // 4 KB
  __shared__ __align__(16) int sTable[C_];              // 512 B

  const int lane = threadIdx.x & 31;
  const int w    = threadIdx.x >> 5;
  const int b    = blockIdx.x * 8 + w;

  // --- async global->LDS staging (one b128 per lane: 32 x 16B = 512B row) ---
  {
    unsigned rowLds = (unsigned)(size_t)&sBest[w][lane * 16];
    uint64_t rowG   = (uint64_t)(size_t)(best + (size_t)b * T_ + lane * 16);
    asm volatile("global_load_async_to_lds_b128 %0, %1, off"
                 :: "v"(rowLds), "v"(rowG) : "memory");
  }
  if (w == 0 && lane < 8) {  // 8 lanes x 16B = 128B: table entries 0..31 only
    unsigned tabLds = (unsigned)(size_t)&sTable[lane * 4];
    uint64_t tabG   = (uint64_t)(size_t)(table + lane * 4);
    asm volatile("global_load_async_to_lds_b128 %0, %1, off"
                 :: "v"(tabLds), "v"(tabG) : "memory");
  }
  asm volatile("s_wait_asynccnt 0x0" ::: "memory");
  __syncthreads();

  const int defc = *defc_p;

  // --- wave32 scan: drop repeats and blanks, left-pack in place in LDS ---
  int carry = -1;  // prev label before t=0 is -1 (reference convention)
  int cnt = 0;     // number of packed labels so far (uniform across lanes)
#pragma unroll
  for (int chunk = 0; chunk < T_ / 32; ++chunk) {
    const int t   = chunk * 32 + lane;
    const int bst = (int)sBest[w][t];
    int prev = __builtin_amdgcn_ds_bpermute((lane << 2) - 4, bst);
    if (lane == 0) prev = carry;
    carry = __builtin_amdgcn_ds_bpermute(31 << 2, bst);  // lane31 -> next chunk
    const bool valid = (bst != prev) && (bst != (C_ - 1));
    const unsigned m = __builtin_amdgcn_ballot_w32(valid);
    const int pos = cnt + __builtin_popcount(m & ((1u << lane) - 1u));
    // In-place compaction is safe: wave-wide DS read of this chunk precedes
    // the write, and every write pos <= its read index t.
    if (valid) sBest[w][pos] = (unsigned char)bst;
    cnt += __builtin_popcount(m);
  }
  __syncthreads();

  // --- emit: table lookup for packed labels, default_char for padding ---
#pragma unroll
  for (int chunk = 0; chunk < T_ / 32; ++chunk) {
    const int t  = chunk * 32 + lane;
    const int ch = (t < cnt) ? sTable[sBest[w][t]] : defc;
    out[(size_t)b * T_ + t] = (float)ch;  // coalesced 128B per wave store
  }
}

// ---------------------------------------------------------------------------
extern "C" void kernel_launch(void* const* d_in, const int* in_sizes, int n_in,
                              void* d_out, int out_size, void* d_ws, size_t ws_size,
                              hipStream_t stream) {
  (void)in_sizes; (void)n_in; (void)out_size; (void)ws_size;
  const float* probs = (const float*)d_in[0];   // [B,T,C] f32
  const int*   table = (const int*)d_in[1];     // [C] i32
  const int*   defc  = (const int*)d_in[2];     // scalar i32
  float* out = (float*)d_out;                   // [B,T] char codes as f32
  unsigned char* best = (unsigned char*)d_ws;   // [B*T] u8 labels (512 KB)

  ctc_argmax_kernel<<<dim3((B_ * T_) / 256), dim3(256), 0, stream>>>(probs, best);
  ctc_compact_kernel<<<dim3(B_ / 8), dim3(256), 0, stream>>>(best, table, defc, out);
}